// GELU269_23648089932080
// MI455X (gfx1250) — compile-verified
//
#include <hip/hip_runtime.h>
#include <math.h>

// Problem constants from the reference: B=8, T=4096, D=1024, N=512
#define D_DIM   1024
#define N_KEYS  512
#define NB1     512     // blocks for the two streaming kernels
#define TPB     256     // 256 threads * float4 = one 1024-wide row per iteration

typedef __attribute__((ext_vector_type(2))) float v2f;
typedef __attribute__((ext_vector_type(4))) float v4f;
typedef __attribute__((ext_vector_type(8))) float v8f;

__device__ __forceinline__ float gelu_tanh(float x) {
    const float S = 0.7978845608028654f;   // sqrt(2/pi)
    float x3 = x * x * x;
    float z2 = 2.0f * S * (x + 0.044715f * x3);
    float e  = __expf(z2);                 // v_exp_f32
    float th = 1.0f - 2.0f / (e + 1.0f);   // tanh with correct +-1 saturation
    return 0.5f * x * (1.0f + th);
}

// ---------------------------------------------------------------------------
// Kernel 1: per-block column partial sums of gelu(x). Deterministic (no float
// atomics): block b writes its 1024 partial sums to partials[b*1024 + d].
// ---------------------------------------------------------------------------
__global__ void colsum_gelu_kernel(const float* __restrict__ x,
                                   float* __restrict__ partials,
                                   int rows) {
    const int tid = threadIdx.x;           // 0..255 -> columns 4*tid..4*tid+3
    const int b   = blockIdx.x;
    const v4f* xv = (const v4f*)x;
    float a0 = 0.f, a1 = 0.f, a2 = 0.f, a3 = 0.f;
    for (int r = b; r < rows; r += NB1) {
        v4f v = xv[(size_t)r * (D_DIM / 4) + tid];
        a0 += gelu_tanh(v.x);
        a1 += gelu_tanh(v.y);
        a2 += gelu_tanh(v.z);
        a3 += gelu_tanh(v.w);
    }
    float* p = partials + (size_t)b * D_DIM + 4 * tid;
    p[0] = a0; p[1] = a1; p[2] = a2; p[3] = a3;
}

// ---------------------------------------------------------------------------
// Kernel 2: single 512-thread block (16 waves).
//  A) reduce partials -> m_curr, normalize -> q (LDS)
//  B) sims = keys @ q via V_WMMA_F32_16X16X4_F32 (q broadcast across N=16)
//  C) masked softmax + first-index argmax
//  D) retrieved = attn @ vals ; delta = retrieved - global_sum/n_global ; t
// ---------------------------------------------------------------------------
__global__ void retrieval_kernel(const float* __restrict__ partials,
                                 const float* __restrict__ log_beta,
                                 const float* __restrict__ log_k_inject,
                                 const float* __restrict__ keys,
                                 const float* __restrict__ vals,
                                 const float* __restrict__ gsum,
                                 const unsigned char* __restrict__ mask,
                                 const int* __restrict__ hit_count,
                                 const int* __restrict__ n_global,
                                 float* __restrict__ delta_out,
                                 float* __restrict__ t_out,
                                 int rows) {
    __shared__ float q_lds[D_DIM];
    __shared__ float sims[N_KEYS];
    __shared__ float logit[N_KEYS];
    __shared__ float attn[N_KEYS];
    __shared__ float redv[N_KEYS];
    __shared__ int   redi[N_KEYS];

    const int tid = threadIdx.x;           // 0..511

    // --- Phase A: column means of y ---
    float m0 = 0.f, m1 = 0.f;
    for (int bb = 0; bb < NB1; ++bb) {
        m0 += partials[(size_t)bb * D_DIM + tid];
        m1 += partials[(size_t)bb * D_DIM + tid + 512];
    }
    const float inv_rows = 1.0f / (float)rows;
    m0 *= inv_rows;  m1 *= inv_rows;

    // --- Phase A2: L2 norm of m_curr ---
    redv[tid] = m0 * m0 + m1 * m1;
    __syncthreads();
    for (int s = 256; s > 0; s >>= 1) {
        if (tid < s) redv[tid] += redv[tid + s];
        __syncthreads();
    }
    const float inorm = 1.0f / fmaxf(sqrtf(redv[0]), 1e-12f);
    __syncthreads();
    q_lds[tid]       = m0 * inorm;
    q_lds[tid + 512] = m1 * inorm;
    __syncthreads();

    // --- Phase B: sims = keys @ q via WMMA f32 16x16x4 ---
    // A-matrix (16x4 f32): lanes 0-15 hold {K=0,K=1}, lanes 16-31 hold {K=2,K=3}
    // B-matrix (4x16): q[k] broadcast across all 16 columns -> every output
    // column of D equals dot(keys_row, q). Accumulate over 256 K-tiles.
    {
        const int wave = tid >> 5;
        const int lane = tid & 31;
        const int m    = lane & 15;
        const int kh   = (lane >> 4) << 1;          // 0 or 2
        for (int tile = wave; tile < N_KEYS / 16; tile += 16) {
            const int row0 = tile * 16;
            const float* krow = keys + (size_t)(row0 + m) * D_DIM + kh;
            v8f c = (v8f)0.0f;
            for (int k0 = 0; k0 < D_DIM; k0 += 4) {
                v2f a, b;
                a.x = krow[k0];
                a.y = krow[k0 + 1];
                b.x = q_lds[k0 + kh];
                b.y = q_lds[k0 + kh + 1];
                c = __builtin_amdgcn_wmma_f32_16x16x4_f32(
                        /*neg_a=*/false, a, /*neg_b=*/false, b,
                        /*c_mod=*/(short)0, c,
                        /*reuse_a=*/false, /*reuse_b=*/false);
            }
            // C/D layout: VGPR r, lanes 0-15 -> M=r (N=lane); lanes 16-31 -> M=8+r
            if (lane == 0) {
                #pragma unroll
                for (int r = 0; r < 8; ++r) sims[row0 + r] = c[r];
            } else if (lane == 16) {
                #pragma unroll
                for (int r = 0; r < 8; ++r) sims[row0 + 8 + r] = c[r];
            }
        }
    }
    __syncthreads();

    // --- Phase C: masked softmax over 512 + first-index argmax ---
    const float beta = fminf(fmaxf(__expf(log_beta[0]), 1.0f), 50.0f);
    const float lg   = mask[tid] ? beta * sims[tid] : -1e9f;
    logit[tid] = lg;
    redv[tid]  = lg;
    redi[tid]  = tid;
    __syncthreads();
    for (int s = 256; s > 0; s >>= 1) {
        if (tid < s) {
            float vo = redv[tid], vn = redv[tid + s];
            if (vn > vo || (vn == vo && redi[tid + s] < redi[tid])) {
                redv[tid] = vn;
                redi[tid] = redi[tid + s];
            }
        }
        __syncthreads();
    }
    const float mx      = redv[0];
    const int   nearest = redi[0];
    __syncthreads();
    const float e = __expf(logit[tid] - mx);
    attn[tid] = e;
    redv[tid] = e;
    __syncthreads();
    for (int s = 256; s > 0; s >>= 1) {
        if (tid < s) redv[tid] += redv[tid + s];
        __syncthreads();
    }
    const float inv_den = 1.0f / redv[0];
    __syncthreads();
    attn[tid] *= inv_den;
    __syncthreads();

    // --- Phase D: retrieved = attn @ vals ; delta ; t ---
    float acc0 = 0.f, acc1 = 0.f;
    for (int n = 0; n < N_KEYS; ++n) {
        const float an = attn[n];
        acc0 += an * vals[(size_t)n * D_DIM + tid];
        acc1 += an * vals[(size_t)n * D_DIM + tid + 512];
    }
    int ngv = n_global[0]; if (ngv < 1) ngv = 1;
    const float ing = 1.0f / (float)ngv;
    delta_out[tid]       = acc0 - gsum[tid] * ing;
    delta_out[tid + 512] = acc1 - gsum[tid + 512] * ing;

    if (tid == 0) {
        const float sp  = mask[nearest] ? sims[nearest] : -1.0f;
        const int   hit = hit_count[nearest] + (sp > 0.85f ? 1 : 0);
        const float kin = fminf(fmaxf(__expf(log_k_inject[0]), 0.001f), 2.0f);
        t_out[0] = kin * __logf((float)hit + 1.0f);
    }
}

// ---------------------------------------------------------------------------
// Kernel 3: out = gelu(x) + t * delta[d]   (streaming, float4)
// ---------------------------------------------------------------------------
__global__ void apply_kernel(const float* __restrict__ x,
                             const float* __restrict__ delta,
                             const float* __restrict__ t_ptr,
                             float* __restrict__ out,
                             int rows) {
    const int tid = threadIdx.x;
    const int b   = blockIdx.x;
    const float t = t_ptr[0];
    const v4f* dv = (const v4f*)delta;
    v4f td = dv[tid];
    td.x *= t; td.y *= t; td.z *= t; td.w *= t;

    const v4f* xv = (const v4f*)x;
    v4f*       ov = (v4f*)out;
    for (int r = b; r < rows; r += NB1) {
        const size_t idx = (size_t)r * (D_DIM / 4) + tid;
        v4f v = xv[idx];
        v4f o;
        o.x = gelu_tanh(v.x) + td.x;
        o.y = gelu_tanh(v.y) + td.y;
        o.z = gelu_tanh(v.z) + td.z;
        o.w = gelu_tanh(v.w) + td.w;
        ov[idx] = o;
    }
}

// ---------------------------------------------------------------------------
extern "C" void kernel_launch(void* const* d_in, const int* in_sizes, int n_in,
                              void* d_out, int out_size, void* d_ws, size_t ws_size,
                              hipStream_t stream) {
    const float*         x        = (const float*)d_in[0];
    const float*         log_beta = (const float*)d_in[1];
    const float*         log_kinj = (const float*)d_in[2];
    const float*         keys     = (const float*)d_in[3];
    const float*         vals     = (const float*)d_in[4];
    const float*         gsum     = (const float*)d_in[5];
    const unsigned char* mask     = (const unsigned char*)d_in[6];
    const int*           hitc     = (const int*)d_in[7];
    const int*           nglob    = (const int*)d_in[8];

    const int rows = in_sizes[0] / D_DIM;   // B*T = 32768

    float* ws       = (float*)d_ws;
    float* partials = ws;                          // NB1 * 1024 floats (2 MiB)
    float* delta    = ws + (size_t)NB1 * D_DIM;    // 1024 floats
    float* tval     = delta + D_DIM;               // 1 float

    colsum_gelu_kernel<<<NB1, TPB, 0, stream>>>(x, partials, rows);
    retrieval_kernel<<<1, 512, 0, stream>>>(partials, log_beta, log_kinj,
                                            keys, vals, gsum, mask, hitc, nglob,
                                            delta, tval, rows);
    apply_kernel<<<NB1, TPB, 0, stream>>>(x, delta, tval, (float*)d_out, rows);
}